// IDM_SGC_topk_52733608461003
// MI455X (gfx1250) — compile-verified
//
#include <hip/hip_runtime.h>
#include <math.h>

typedef __attribute__((ext_vector_type(2))) float v2f;
typedef __attribute__((ext_vector_type(8))) float v8f;

#define MDIM 512
#define NDIM 200000
#define KDIM 100
#define GAMMA_C 0.8f
#define EPS_C 1e-12f
#define NSWEEPS 10
#define NCHUNK 2000          // 100 chunks * 2000 = 200000, multiple of 4
#define NCHUNKS (NDIM / NCHUNK)

// D = A(16x4 f32) * B(4x16 f32) + C(16x16 f32), wave32 striped
__device__ __forceinline__ v8f wmma_f32_4(v2f a, v2f b, v8f c) {
  return __builtin_amdgcn_wmma_f32_16x16x4_f32(
      /*neg_a=*/false, a, /*neg_b=*/false, b,
      /*c_mod=*/(short)0, c, /*reuse_a=*/false, /*reuse_b=*/false);
}

// ---------------------------------------------------------------- init
__global__ void idm_zero_init(float* __restrict__ Y, float* __restrict__ norm2) {
  int idx = blockIdx.x * blockDim.x + threadIdx.x;
  if (idx < MDIM * KDIM) Y[idx] = 0.0f;
  if (idx == 0) *norm2 = 0.0f;
}

// ---------------------------------------------------------------- FF = F^T F
__global__ void idm_syrk(const float* __restrict__ F, float* __restrict__ FF) {
  int j = blockIdx.x * blockDim.x + threadIdx.x;
  int i = blockIdx.y * blockDim.y + threadIdx.y;
  if (i < MDIM && j < MDIM) {
    float s = 0.0f;
    for (int l = 0; l < MDIM; ++l) s += F[l * MDIM + i] * F[l * MDIM + j];
    FF[i * MDIM + j] = s;
  }
}

// ---------------------------------------------------------------- ||FF||_F^2
__global__ void idm_frob(const float* __restrict__ FF, float* __restrict__ norm2) {
  int idx = blockIdx.x * blockDim.x + threadIdx.x;
  float v = FF[idx];
  v *= v;
  for (int off = 16; off > 0; off >>= 1) v += __shfl_xor(v, off, 32);
  if ((threadIdx.x & 31) == 0) atomicAdd(norm2, v);
}

// ---------------------------------------------------------------- A = gF, V = I
__global__ void idm_scale_init(const float* __restrict__ FF, const float* __restrict__ norm2,
                               float* __restrict__ A, float* __restrict__ V) {
  int idx = blockIdx.x * blockDim.x + threadIdx.x;
  float inv = 1.0f / (sqrtf(*norm2) + EPS_C);
  A[idx] = FF[idx] * inv;
  int i = idx / MDIM, j = idx % MDIM;
  V[idx] = (i == j) ? 1.0f : 0.0f;
}

// ---------------------------------------------------------------- eigh via parallel Jacobi
__global__ __launch_bounds__(512) void idm_jacobi(float* __restrict__ A, float* __restrict__ V,
                                                  float* __restrict__ lam) {
  __shared__ float cs[256], sn[256];
  __shared__ int pp[256], qq[256];
  const int tid = threadIdx.x;
  const int bdim = blockDim.x;
  for (int sweep = 0; sweep < NSWEEPS; ++sweep) {
    for (int r = 0; r < MDIM - 1; ++r) {
      if (tid < 256) {
        int p, q;
        if (tid == 0) { p = MDIM - 1; q = r % (MDIM - 1); }
        else {
          p = (r + tid) % (MDIM - 1);
          q = (r + (MDIM - 1) - tid) % (MDIM - 1);
        }
        if (p > q) { int t = p; p = q; q = t; }
        float app = A[p * MDIM + p], aqq = A[q * MDIM + q], apq = A[p * MDIM + q];
        float c = 1.0f, s = 0.0f;
        if (fabsf(apq) > 1e-20f) {
          float tau = (aqq - app) / (2.0f * apq);
          float t = copysignf(1.0f, tau) / (fabsf(tau) + sqrtf(1.0f + tau * tau));
          c = rsqrtf(1.0f + t * t);
          s = t * c;
        }
        pp[tid] = p; qq[tid] = q; cs[tid] = c; sn[tid] = s;
      }
      __syncthreads();
      // row phase: A <- J^T A  (pairs have disjoint rows -> no intra-phase hazard)
      for (int pr = 0; pr < 256; ++pr) {
        int p = pp[pr], q = qq[pr];
        float c = cs[pr], s = sn[pr];
        for (int j = tid; j < MDIM; j += bdim) {
          float ap = A[p * MDIM + j], aq = A[q * MDIM + j];
          A[p * MDIM + j] = c * ap - s * aq;
          A[q * MDIM + j] = s * ap + c * aq;
        }
      }
      __syncthreads();
      // col phase: A <- A J, V <- V J  (disjoint columns)
      for (int pr = 0; pr < 256; ++pr) {
        int p = pp[pr], q = qq[pr];
        float c = cs[pr], s = sn[pr];
        for (int j = tid; j < MDIM; j += bdim) {
          float ap = A[j * MDIM + p], aq = A[j * MDIM + q];
          A[j * MDIM + p] = c * ap - s * aq;
          A[j * MDIM + q] = s * ap + c * aq;
          float vp = V[j * MDIM + p], vq = V[j * MDIM + q];
          V[j * MDIM + p] = c * vp - s * vq;
          V[j * MDIM + q] = s * vp + c * vq;
        }
      }
      __syncthreads();
    }
  }
  for (int i = tid; i < MDIM; i += bdim) lam[i] = A[i * MDIM + i];
}

// ---------------------------------------------------------------- Y = X @ Q_S (split-K, WMMA f32)
// Tail k-tile handled with clamped-address loads + 0/1 mask multiply:
// no exec-mask divergence anywhere in the streaming loop.
__global__ __launch_bounds__(32) void idm_gemm_xqs(const float* __restrict__ X,
                                                   const float* __restrict__ Qs,
                                                   float* __restrict__ Y) {
  const int mt = blockIdx.x;                 // 32 M-tiles
  const int chunk = blockIdx.y;              // NCHUNKS n-chunks
  const int lane = threadIdx.x;
  const int hi = lane >> 4, lo = lane & 15;
  const int m0 = mt * 16;
  const long n0 = (long)chunk * NCHUNK;
  const long n1 = n0 + NCHUNK;

  v8f acc[7];
  for (int t = 0; t < 7; ++t)
    for (int r = 0; r < 8; ++r) acc[t][r] = 0.0f;

  // per-lane column index (clamped) and validity mask, uniform across n-loop
  int colc[7];
  float msk[7];
  for (int t = 0; t < 7; ++t) {
    int c = t * 16 + lo;
    msk[t] = (c < KDIM) ? 1.0f : 0.0f;
    colc[t] = (c < KDIM) ? c : 0;
  }

  const float* Xrow = X + (long)(m0 + lo) * NDIM;
  for (long n = n0; n < n1; n += 4) {
    v2f a;                                   // A = X[m0..m0+15, n..n+3]
    a.x = Xrow[n + 2 * hi];
    a.y = Xrow[n + 2 * hi + 1];
    const float* q0 = Qs + (n + 2 * hi) * KDIM;
    for (int t = 0; t < 7; ++t) {
      v2f b;                                 // B = Q_S[n..n+3, t*16..t*16+15]
      b.x = q0[colc[t]] * msk[t];
      b.y = q0[KDIM + colc[t]] * msk[t];
      acc[t] = wmma_f32_4(a, b, acc[t]);
    }
  }
  for (int t = 0; t < 7; ++t) {
    int col = t * 16 + lo;
    if (col < KDIM)
      for (int r = 0; r < 8; ++r)
        atomicAdd(&Y[(m0 + r + 8 * hi) * KDIM + col], acc[t][r]);
  }
}

// ---------------------------------------------------------------- U = G .* (Q_F^T @ Y)
__global__ void idm_proj_gate(const float* __restrict__ V, const float* __restrict__ Y,
                              const float* __restrict__ lamF, const float* __restrict__ lamS,
                              float* __restrict__ U) {
  int i = blockIdx.y;
  int k = blockIdx.x * blockDim.x + threadIdx.x;
  if (k < KDIM) {
    float s = 0.0f;
    for (int l = 0; l < MDIM; ++l) s += V[l * MDIM + i] * Y[l * KDIM + k];
    float g = 1.0f / (1.0f - GAMMA_C * lamF[i] * lamS[k]);
    U[i * KDIM + k] = s * g;
  }
}

// ---------------------------------------------------------------- W = Q_F @ U
__global__ void idm_backproj(const float* __restrict__ V, const float* __restrict__ U,
                             float* __restrict__ W) {
  int i = blockIdx.y;
  int k = blockIdx.x * blockDim.x + threadIdx.x;
  if (k < KDIM) {
    float s = 0.0f;
    for (int l = 0; l < MDIM; ++l) s += V[i * MDIM + l] * U[l * KDIM + k];
    W[i * KDIM + k] = s;
  }
}

// ---------------------------------------------------------------- Z = W @ Q_S^T (WMMA f32)
__global__ __launch_bounds__(128) void idm_gemm_out(const float* __restrict__ W,
                                                    const float* __restrict__ Qs,
                                                    float* __restrict__ Z) {
  const int wave = threadIdx.x >> 5;         // 4 waves -> 4 N-tiles
  const int lane = threadIdx.x & 31;
  const int hi = lane >> 4, lo = lane & 15;
  const int m0 = blockIdx.y * 16;
  const long j0 = (long)blockIdx.x * 64 + wave * 16;

  v8f acc;
  for (int r = 0; r < 8; ++r) acc[r] = 0.0f;

  const float* Wrow = W + (m0 + lo) * KDIM;  // A = W[m0..m0+15, kk..kk+3]
  const float* Qrow = Qs + (j0 + lo) * KDIM; // B = Q_S^T[kk..kk+3, j0..j0+15]
  for (int kk = 0; kk < KDIM; kk += 4) {     // 100 % 4 == 0 -> 25 WMMAs, no tail
    v2f a, b;
    a.x = Wrow[kk + 2 * hi];
    a.y = Wrow[kk + 2 * hi + 1];
    b.x = Qrow[kk + 2 * hi];
    b.y = Qrow[kk + 2 * hi + 1];
    acc = wmma_f32_4(a, b, acc);
  }
  for (int r = 0; r < 8; ++r)
    Z[(long)(m0 + r + 8 * hi) * NDIM + j0 + lo] = acc[r];
}

// ---------------------------------------------------------------- launch
extern "C" void kernel_launch(void* const* d_in, const int* in_sizes, int n_in,
                              void* d_out, int out_size, void* d_ws, size_t ws_size,
                              hipStream_t stream) {
  const float* X    = (const float*)d_in[0];   // 512 x 200000
  const float* F    = (const float*)d_in[1];   // 512 x 512
  const float* Qs   = (const float*)d_in[2];   // 200000 x 100
  const float* lamS = (const float*)d_in[3];   // 100
  float* Z = (float*)d_out;                    // 512 x 200000
  float* ws = (float*)d_ws;

  float* FF    = ws;                            // 262144
  float* A     = FF + MDIM * MDIM;              // 262144
  float* V     = A + MDIM * MDIM;               // 262144 (Q_F)
  float* Y     = V + MDIM * MDIM;               // 51200
  float* U     = Y + MDIM * KDIM;               // 51200
  float* W     = U + MDIM * KDIM;               // 51200
  float* lamF  = W + MDIM * KDIM;               // 512
  float* norm2 = lamF + MDIM;                   // 1

  idm_zero_init<<<(MDIM * KDIM + 255) / 256, 256, 0, stream>>>(Y, norm2);
  idm_syrk<<<dim3(32, 32), dim3(16, 16), 0, stream>>>(F, FF);
  idm_frob<<<(MDIM * MDIM) / 256, 256, 0, stream>>>(FF, norm2);
  idm_scale_init<<<(MDIM * MDIM) / 256, 256, 0, stream>>>(FF, norm2, A, V);
  idm_jacobi<<<1, 512, 0, stream>>>(A, V, lamF);
  idm_gemm_xqs<<<dim3(MDIM / 16, NCHUNKS), 32, 0, stream>>>(X, Qs, Y);
  idm_proj_gate<<<dim3(1, MDIM), 128, 0, stream>>>(V, Y, lamF, lamS, U);
  idm_backproj<<<dim3(1, MDIM), 128, 0, stream>>>(V, U, W);
  idm_gemm_out<<<dim3(NDIM / 64, MDIM / 16), 128, 0, stream>>>(W, Qs, Z);
}